// BoxModule_18056042512998
// MI455X (gfx1250) — compile-verified
//
#include <hip/hip_runtime.h>
#include <math.h>

#define IMW1 511.0f
#define IMH1 511.0f
#define BB_THR 0.5f
#define BB_NUM 100
#define NB 32
#define NN 65536
#define TT 512
#define KK (NN / TT) /* 128 scores per thread, register resident */
#define NWAVES (TT / 32)

#define NEG_INF (-INFINITY)

// ---------------------------------------------------------------------------
// 16B global -> LDS copy via CDNA5 async-to-LDS path (probe-verified lowering).
// ---------------------------------------------------------------------------
typedef int v4i __attribute__((vector_size(4 * sizeof(int))));
typedef __attribute__((address_space(1))) v4i* g_v4i_p;
typedef __attribute__((address_space(3))) v4i* l_v4i_p;

__device__ __forceinline__ void copy16_g2lds(void* lds_dst, const void* gsrc) {
#if __has_builtin(__builtin_amdgcn_global_load_async_to_lds_b128) && \
    __has_builtin(__builtin_amdgcn_s_wait_asynccnt)
  __builtin_amdgcn_global_load_async_to_lds_b128(
      (g_v4i_p)(v4i*)(void*)(size_t)gsrc,  // strip const, generic->AS1
      (l_v4i_p)(v4i*)lds_dst,              // generic->AS3
      /*offset=*/0, /*cpol=*/0);
  __builtin_amdgcn_s_wait_asynccnt(0);
#else
  *(float4*)lds_dst = *(const float4*)gsrc;
#endif
}

// ---------------------------------------------------------------------------
// Kernel 1: decode.  scores = softmax(cl)[...,0]; boxes = clamped anc +/- re.
// Pure streaming pass, memory bound (~90MB total @ 23.3 TB/s).
// ---------------------------------------------------------------------------
__global__ void decode_kernel(const float* __restrict__ cl,
                              const float* __restrict__ re,
                              const float* __restrict__ anc,
                              float* __restrict__ boxes,
                              float* __restrict__ scores) {
  int gid = blockIdx.x * blockDim.x + threadIdx.x;  // over B*N
  if (gid >= NB * NN) return;
  int i = gid & (NN - 1);

  float2 c = ((const float2*)cl)[gid];
  float4 r = ((const float4*)re)[gid];
  float4 a = ((const float4*)anc)[i];

  // numerically match jax.nn.softmax (subtract max, exp, normalize)
  float mx = fmaxf(c.x, c.y);
  float e0 = expf(c.x - mx);
  float e1 = expf(c.y - mx);
  scores[gid] = e0 / (e0 + e1);

  float4 bb;
  bb.x = fmaxf(a.x - r.x, 0.0f);
  bb.y = fmaxf(a.y - r.y, 0.0f);
  bb.z = fminf(a.z + r.z, IMW1);
  bb.w = fminf(a.w + r.w, IMH1);
  ((float4*)boxes)[gid] = bb;
}

// ---------------------------------------------------------------------------
// Kernel 2: greedy NMS, one workgroup (16 wave32 waves) per image.
// Scores live in registers (128/lane) for all 100 iterations; boxes stream
// from the 192MB L2 each iteration (32MB total working set).  Per-thread
// argmax is maintained incrementally: a rescan is only needed when the
// thread's own current argmax entry got suppressed.
// ---------------------------------------------------------------------------
__global__ __launch_bounds__(TT) void nms_kernel(
    const float* __restrict__ boxes, const float* __restrict__ scores,
    float* __restrict__ out_bb, float* __restrict__ out_ff) {
  const int b = blockIdx.x;
  const int t = threadIdx.x;

  const float4* __restrict__ boxb = (const float4*)(boxes + (size_t)b * NN * 4);
  const float* __restrict__ sb = scores + (size_t)b * NN;

  __shared__ float red_v[32];
  __shared__ int red_i[32];
  __shared__ __align__(16) float selbox[4];
  __shared__ int sel_i;
  __shared__ int sel_valid;

  // Register-resident working scores; thread t owns indices {t, t+TT, ...}
  float s[KK];
#pragma unroll
  for (int k = 0; k < KK; ++k) s[k] = sb[t + k * TT];

  // warm the L2/WGP path for the box array (gfx1250 global_prefetch_b8)
  __builtin_prefetch(boxb + t, 0, 3);

  // sentinel slots for unused wave positions in the final reduction
  if (t < 32) {
    red_v[t] = NEG_INF;
    red_i[t] = 0x7fffffff;
  }
  __syncthreads();

  // leader-only carry state (meaningful in thread 0 only)
  int prev_idx = 0;
  float cur_ff = 0.0f;
  float cbx = 0.0f, cby = 0.0f, cbz = 0.0f, cbw = 0.0f;

  // incrementally maintained per-thread argmax
  float tm = NEG_INF;
  int tmi = t;
  bool dirty = true;

  for (int it = 0; it < BB_NUM; ++it) {
    // ---- Phase A: per-thread argmax (first-index tie break) ----
    if (dirty) {
      tm = NEG_INF;
      tmi = t;
#pragma unroll
      for (int k = 0; k < KK; ++k) {
        float v = s[k];
        int idx = t + k * TT;  // ascending within thread -> keep-first on ties
        bool g = v > tm;
        tm = g ? v : tm;
        tmi = g ? idx : tmi;
      }
      dirty = false;
    }
    float m = tm;
    int mi = tmi;
    // wave32 reduction: bigger score wins, equal score -> smaller index
#pragma unroll
    for (int off = 16; off > 0; off >>= 1) {
      float om = __shfl_xor(m, off, 32);
      int oi = __shfl_xor(mi, off, 32);
      bool take = (om > m) || ((om == m) && (oi < mi));
      m = take ? om : m;
      mi = take ? oi : mi;
    }
    if ((t & 31) == 0) {
      red_v[t >> 5] = m;
      red_i[t >> 5] = mi;
    }
    __syncthreads();

    if (t < 32) {
      m = red_v[t];
      mi = red_i[t];
#pragma unroll
      for (int off = 16; off > 0; off >>= 1) {
        float om = __shfl_xor(m, off, 32);
        int oi = __shfl_xor(mi, off, 32);
        bool take = (om > m) || ((om == m) && (oi < mi));
        m = take ? om : m;
        mi = take ? oi : mi;
      }
      if (t == 0) {
        bool valid = (m > NEG_INF);
        int idx = valid ? mi : prev_idx;
        sel_i = idx;
        sel_valid = valid ? 1 : 0;
        if (valid) {
          prev_idx = idx;
          // async DMA the winning box straight into the LDS broadcast slot
          copy16_g2lds(selbox, (const void*)(boxb + idx));
          cur_ff = m;
          cbx = selbox[0];
          cby = selbox[1];
          cbz = selbox[2];
          cbw = selbox[3];
        }
        out_ff[b * BB_NUM + it] = cur_ff;
        float* ob = out_bb + ((size_t)b * BB_NUM + it) * 4;
        ob[0] = cbx;
        ob[1] = cby;
        ob[2] = cbz;
        ob[3] = cbw;
      }
    }
    __syncthreads();

    // ---- Phase B: suppression (branchless, streams boxes from L2) ----
    if (sel_valid) {
      const float X1 = selbox[0], Y1 = selbox[1], X2 = selbox[2], Y2 = selbox[3];
      const float A = (X2 - X1) * (Y2 - Y1);
      const int idx = sel_i;
#pragma unroll
      for (int k = 0; k < KK; ++k) {
        int i = t + k * TT;
        float4 bx = boxb[i];
        float iw = fmaxf(fminf(bx.z, X2) - fmaxf(bx.x, X1), 0.0f);
        float ih = fmaxf(fminf(bx.w, Y2) - fmaxf(bx.y, Y1), 0.0f);
        float inter = iw * ih;
        float ai = (bx.z - bx.x) * (bx.w - bx.y);
        float iou = inter / (ai + A - inter + 1e-9f);
        bool kill = (iou > BB_THR) || (i == idx);  // explicit self-suppress
        // killing a non-argmax entry cannot change this thread's argmax
        dirty = dirty || (kill && (i == tmi));
        s[k] = kill ? NEG_INF : s[k];
      }
    }
    // no extra barrier needed: next iteration's red[] writes happen before
    // barrier1', and sel_* is rewritten by t0 only after barrier1'.
  }
}

// ---------------------------------------------------------------------------
extern "C" void kernel_launch(void* const* d_in, const int* in_sizes, int n_in,
                              void* d_out, int out_size, void* d_ws,
                              size_t ws_size, hipStream_t stream) {
  const float* cl = (const float*)d_in[0];   // (B,N,2)
  const float* re = (const float*)d_in[1];   // (B,N,4)
  const float* anc = (const float*)d_in[2];  // (N,4)

  float* boxes = (float*)d_ws;                    // B*N*4 floats (32MB)
  float* scores = boxes + (size_t)NB * NN * 4;    // B*N floats   (8MB)

  float* out_bb = (float*)d_out;                  // B*100*4
  float* out_ff = out_bb + (size_t)NB * BB_NUM * 4;

  int total = NB * NN;
  decode_kernel<<<total / 256, 256, 0, stream>>>(cl, re, anc, boxes, scores);
  nms_kernel<<<NB, TT, 0, stream>>>(boxes, scores, out_bb, out_ff);
}